// RoiPoolingConv_7378753815348
// MI455X (gfx1250) — compile-verified
//
#include <hip/hip_runtime.h>
#include <stdint.h>

// ROI-Align (bilinear 7x7 pooling) for img (1,256,256,512) fp32 NHWC, 300 ROIs.
// Bandwidth-bound kernel: uses gfx1250 async Global->LDS DMA (ASYNCcnt path)
// as a deep per-wave prefetch pipeline, drained in order with s_wait_asynccnt.
// Output is streamed with non-temporal B128 stores so the (L2-resident) image
// keeps its residency in the 192MB L2.

#define POOL    7
#define CH      512
#define IMG_H   256
#define IMG_W   256
#define THREADS 128   // 4 waves; each thread owns 4 channels (one float4)

typedef float v4f __attribute__((ext_vector_type(4)));   // native vector for NT store

__device__ __forceinline__ void async_copy_b128(uint64_t gaddr, uint32_t lds_addr) {
    // GLOBAL_LOAD_ASYNC_TO_LDS_B128: per-lane 16B global -> LDS, tracked by ASYNCcnt
    asm volatile("global_load_async_to_lds_b128 %0, %1, off"
                 :: "v"(lds_addr), "v"(gaddr)
                 : "memory");
}

__global__ __launch_bounds__(THREADS)
void roi_align_async_kernel(const float* __restrict__ img,
                            const int*   __restrict__ rois,
                            float*       __restrict__ out)
{
    // 7 px cells x 4 corners x 128 lanes x 16B = 57344 B staging buffer
    __shared__ v4f smem[POOL * 4 * THREADS];

    const int blk = blockIdx.x;          // roi * 7 + py
    const int roi = blk / POOL;
    const int py  = blk - roi * POOL;
    const int tid = threadIdx.x;

    const int rx = rois[roi * 4 + 0];
    const int ry = rois[roi * 4 + 1];
    const int rw = rois[roi * 4 + 2];
    const int rh = rois[roi * 4 + 3];

    // ---- y-axis coords for this pooled row (matches reference _axis_coords) ----
    float fy  = (py + 0.5f) * (float)rh / (float)POOL - 0.5f;
    float fy0 = floorf(fy);
    float ty  = fy - fy0;
    int   ay  = (int)fy0;
    int y0 = min(max(ay,     0), rh - 1) + ry;
    int y1 = min(max(ay + 1, 0), rh - 1) + ry;
    y0 = min(max(y0, 0), IMG_H - 1);
    y1 = min(max(y1, 0), IMG_H - 1);

    const uint64_t img_base = (uint64_t)(uintptr_t)img;
    // Generic pointer to LDS: low 32 bits are the wave-relative LDS byte address.
    const uint32_t lds_base  = (uint32_t)(uintptr_t)(&smem[0]);
    const uint32_t lane_byte = (uint32_t)tid * 16u;

    float txs[POOL];

    // ---- issue phase: queue all 28 async B128 transfers (deep prefetch) ----
    #pragma unroll
    for (int px = 0; px < POOL; ++px) {
        float fx  = (px + 0.5f) * (float)rw / (float)POOL - 0.5f;
        float fx0 = floorf(fx);
        txs[px]   = fx - fx0;
        int   ax  = (int)fx0;
        int x0 = min(max(ax,     0), rw - 1) + rx;
        int x1 = min(max(ax + 1, 0), rw - 1) + rx;
        x0 = min(max(x0, 0), IMG_W - 1);
        x1 = min(max(x1, 0), IMG_W - 1);

        const uint64_t cbyte = (uint64_t)tid * 16u;  // channel byte offset
        uint64_t g_tl = img_base + ((uint64_t)(y0 * IMG_W + x0) * (CH * 4)) + cbyte;
        uint64_t g_tr = img_base + ((uint64_t)(y0 * IMG_W + x1) * (CH * 4)) + cbyte;
        uint64_t g_bl = img_base + ((uint64_t)(y1 * IMG_W + x0) * (CH * 4)) + cbyte;
        uint64_t g_br = img_base + ((uint64_t)(y1 * IMG_W + x1) * (CH * 4)) + cbyte;

        uint32_t l = lds_base + (uint32_t)(px * 4) * (THREADS * 16) + lane_byte;
        async_copy_b128(g_tl, l + 0u * (THREADS * 16));
        async_copy_b128(g_tr, l + 1u * (THREADS * 16));
        async_copy_b128(g_bl, l + 2u * (THREADS * 16));
        async_copy_b128(g_br, l + 3u * (THREADS * 16));
    }

    const int out_base = ((roi * POOL + py) * POOL) * CH + tid * 4;

    // ---- drain phase: async loads complete in issue order per wave ----
    #pragma unroll
    for (int px = 0; px < POOL; ++px) {
        switch (px) {  // constant after full unroll; wait for first 4*(px+1) done
            case 0:  asm volatile("s_wait_asynccnt 24" ::: "memory"); break;
            case 1:  asm volatile("s_wait_asynccnt 20" ::: "memory"); break;
            case 2:  asm volatile("s_wait_asynccnt 16" ::: "memory"); break;
            case 3:  asm volatile("s_wait_asynccnt 12" ::: "memory"); break;
            case 4:  asm volatile("s_wait_asynccnt 8"  ::: "memory"); break;
            case 5:  asm volatile("s_wait_asynccnt 4"  ::: "memory"); break;
            default: asm volatile("s_wait_asynccnt 0"  ::: "memory"); break;
        }
        v4f tl = smem[(px * 4 + 0) * THREADS + tid];
        v4f tr = smem[(px * 4 + 1) * THREADS + tid];
        v4f bl = smem[(px * 4 + 2) * THREADS + tid];
        v4f br = smem[(px * 4 + 3) * THREADS + tid];

        const float tx = txs[px];
        v4f top = tl + (tr - tl) * tx;
        v4f bot = bl + (br - bl) * tx;
        v4f o   = top + (bot - top) * ty;

        // Stream-out: output is write-once, never re-read -> non-temporal (TH_NT)
        __builtin_nontemporal_store(o, (v4f*)(out + out_base + px * CH));
    }
}

extern "C" void kernel_launch(void* const* d_in, const int* in_sizes, int n_in,
                              void* d_out, int out_size, void* d_ws, size_t ws_size,
                              hipStream_t stream) {
    const float* img  = (const float*)d_in[0];
    const int*   rois = (const int*)d_in[1];
    float*       out  = (float*)d_out;

    const int n_rois = in_sizes[1] / 4;              // 300
    dim3 grid((unsigned)(n_rois * POOL));            // one block per (roi, pooled row)
    roi_align_async_kernel<<<grid, THREADS, 0, stream>>>(img, rois, out);
}